// LGCProp_66400194396297
// MI455X (gfx1250) — compile-verified
//
#include <hip/hip_runtime.h>
#include <stdint.h>

// LightGCN propagation, MI455X (gfx1250, wave32).
// Pipeline (all on `stream`, graph-capture safe, no host sync):
//   1) zero degree/cursor arrays (ws is poisoned once, never re-poisoned)
//   2) count degrees (int atomics, cold)
//   3) exclusive scan -> CSR row pointers (single-block LDS scan, tiny)
//   4) deg -> 1/sqrt(deg) in place
//   5) build CSR adjacency (cursor atomics, cold)
//   6) init: S = inv*x, out = 0.25*x
//   7) 3 layers x 2 atomic-free gather kernels (hot path):
//        wave-per-node, float4-per-lane, async-LDS staged indices
//        (global_load_async_to_lds_b128 + s_wait_asynccnt), unroll-by-4
//        gather with 4 b128 loads in flight, fused epilogue
//        (out += 0.25*inv*sum ; Snext = inv^2*sum)
//
// Workspace (4B words): Su[nU] | Si0[nI] | Si1[nI] | deg/inv[u+i] |
//                       row_u[u+1] | row_i[i+1] | cur[u+i] | col_u[E+4] | col_i[E+4]
// ~52 MB for the given sizes.

#define D_FEAT 128
#define V4N    (D_FEAT / 4)   // float4 per node row == 32 == wave32 lanes
#define WPB    8              // waves per 256-thread block
#define CAP    64             // neighbor indices staged in LDS per wave

typedef unsigned int u32;

__global__ __launch_bounds__(256) void k_zero_u32(u32* __restrict__ p, int n) {
  int i = blockIdx.x * 256 + threadIdx.x;
  if (i < n) p[i] = 0u;
}

__global__ __launch_bounds__(256) void k_count_deg(const int* __restrict__ eu,
                                                   const int* __restrict__ ei, int E,
                                                   u32* __restrict__ degu,
                                                   u32* __restrict__ degi) {
  int e = blockIdx.x * 256 + threadIdx.x;
  if (e < E) {
    atomicAdd(&degu[eu[e]], 1u);
    atomicAdd(&degi[ei[e]], 1u);
  }
}

// Single-block exclusive scan (Hillis-Steele chunks of 1024 + running carry).
__global__ __launch_bounds__(1024) void k_scan_excl(const u32* __restrict__ deg,
                                                    u32* __restrict__ row, int n) {
  __shared__ u32 buf[1024];
  const int tid = threadIdx.x;
  u32 carry = 0u;
  for (int base = 0; base < n; base += 1024) {
    u32 v = (base + tid < n) ? deg[base + tid] : 0u;
    buf[tid] = v;
    __syncthreads();
    for (int off = 1; off < 1024; off <<= 1) {
      u32 t = (tid >= off) ? buf[tid - off] : 0u;
      __syncthreads();
      buf[tid] += t;
      __syncthreads();
    }
    u32 incl = buf[tid];
    u32 bsum = buf[1023];
    if (base + tid < n) row[base + tid] = carry + incl - v;
    carry += bsum;
    __syncthreads();
  }
  if (tid == 0) row[n] = carry;
}

__global__ __launch_bounds__(256) void k_deg_to_inv(u32* __restrict__ degbase, int n) {
  int i = blockIdx.x * 256 + threadIdx.x;
  if (i < n) {
    u32 d = degbase[i];
    float f = (d > 0u) ? rsqrtf((float)d) : 0.0f;
    ((float*)degbase)[i] = f;
  }
}

__global__ __launch_bounds__(256) void k_build_csr(const int* __restrict__ eu,
                                                   const int* __restrict__ ei, int E,
                                                   const u32* __restrict__ rowu,
                                                   const u32* __restrict__ rowi,
                                                   u32* __restrict__ curu,
                                                   u32* __restrict__ curi,
                                                   u32* __restrict__ colu,
                                                   u32* __restrict__ coli) {
  int e = blockIdx.x * 256 + threadIdx.x;
  if (e >= E) return;
  int u = eu[e], i = ei[e];
  u32 p = atomicAdd(&curu[u], 1u);
  colu[rowu[u] + p] = (u32)i;
  u32 q = atomicAdd(&curi[i], 1u);
  coli[rowi[i] + q] = (u32)u;
}

// S = inv[node] * x ; out = 0.25 * x   (float4 vectorized)
__global__ __launch_bounds__(256) void k_init(const float* __restrict__ x,
                                              const float* __restrict__ inv,
                                              float* __restrict__ S,
                                              float* __restrict__ out, int nvec) {
  int v = blockIdx.x * 256 + threadIdx.x;
  if (v < nvec) {
    int node = v >> 5;  // 32 float4 per node
    float s = inv[node];
    float4 xv = ((const float4*)x)[v];
    float4 o;  o.x  = 0.25f * xv.x; o.y  = 0.25f * xv.y; o.z  = 0.25f * xv.z; o.w  = 0.25f * xv.w;
    float4 sv; sv.x = s * xv.x;     sv.y = s * xv.y;     sv.z = s * xv.z;     sv.w = s * xv.w;
    ((float4*)out)[v] = o;
    ((float4*)S)[v]   = sv;
  }
}

// Hot kernel: one wave per destination node. Neighbor index list is streamed
// into this wave's private LDS slot with the CDNA5 async copy engine
// (global_load_async_to_lds_b128, synced per-wave via s_wait_asynccnt — no
// barrier needed). Each lane owns a float4 slice of the 128-float row. The
// main loop is unrolled by 4: one ds_load_b128 supplies 4 indices and 4
// independent global_load_b128 gathers are kept in flight, accumulated into
// two register chains. Epilogue fuses output accumulation and next-layer
// source scaling.
__global__ __launch_bounds__(256) void k_gather(const u32* __restrict__ row,
                                                const u32* __restrict__ col,
                                                const float* __restrict__ Ssrc,
                                                const float* __restrict__ invd,
                                                float* __restrict__ outAcc,
                                                float* __restrict__ Snext,
                                                int n_dst, float wl, int write_next) {
  __shared__ __align__(16) u32 stage[WPB][CAP];
  const int wave = threadIdx.x >> 5;
  const int lane = threadIdx.x & 31;
  int node = blockIdx.x * WPB + wave;
  if (node >= n_dst) return;                      // wave-uniform exit, no barriers used
  node = __builtin_amdgcn_readfirstlane(node);    // scalarize row-pointer loads

  const u32 start  = row[node];
  const u32 cnt    = row[node + 1] - start;
  const u32 staged = (cnt < (u32)CAP) ? cnt : (u32)CAP;

  // Async-stage the first `staged` indices: lane l copies ints [4l, 4l+3].
  // (col arrays are padded by +4 words so the rounding-up read is in bounds.)
  if ((u32)(4 * lane) < staged) {
    const u32* gsrc = col + start + 4 * lane;
    u32 laddr = (u32)(uintptr_t)&stage[wave][4 * lane];  // low 32 bits == LDS offset
    asm volatile("global_load_async_to_lds_b128 %0, %1, off"
                 :: "v"(laddr), "v"(gsrc)
                 : "memory");
  }

  const float* base = Ssrc + (size_t)(lane << 2);
  float4 acc0; acc0.x = acc0.y = acc0.z = acc0.w = 0.0f;
  float4 acc1; acc1.x = acc1.y = acc1.z = acc1.w = 0.0f;

  // Per-wave counter: only this wave's async copies need to land. No barrier.
  asm volatile("s_wait_asynccnt 0" ::: "memory");

  const u32 lim = staged & ~3u;   // unrolled region (within staged indices)
  u32 k = 0;
  if (lim) {
    uint4 nid = *(const uint4*)&stage[wave][0];   // ds_load_b128 (broadcast)
    for (; k < lim; k += 4) {
      uint4 id = nid;
      if (k + 4 < lim) {
        nid = *(const uint4*)&stage[wave][k + 4];
        __builtin_prefetch((const void*)(base + (size_t)nid.x * D_FEAT), 0, 1);
      }
      float4 a = *(const float4*)(base + (size_t)id.x * D_FEAT);
      float4 b = *(const float4*)(base + (size_t)id.y * D_FEAT);
      float4 c = *(const float4*)(base + (size_t)id.z * D_FEAT);
      float4 d = *(const float4*)(base + (size_t)id.w * D_FEAT);
      acc0.x += a.x; acc0.y += a.y; acc0.z += a.z; acc0.w += a.w;
      acc1.x += b.x; acc1.y += b.y; acc1.z += b.z; acc1.w += b.w;
      acc0.x += c.x; acc0.y += c.y; acc0.z += c.z; acc0.w += c.w;
      acc1.x += d.x; acc1.y += d.y; acc1.z += d.z; acc1.w += d.w;
    }
  }
  for (; k < cnt; ++k) {          // tail + (rare) deg > CAP overflow
    u32 src = (k < staged) ? stage[wave][k] : col[start + k];
    float4 sv = *(const float4*)(base + (size_t)src * D_FEAT);
    acc0.x += sv.x; acc0.y += sv.y; acc0.z += sv.z; acc0.w += sv.w;
  }

  const float s = invd[node];
  float4 t;
  t.x = s * (acc0.x + acc1.x);
  t.y = s * (acc0.y + acc1.y);
  t.z = s * (acc0.z + acc1.z);
  t.w = s * (acc0.w + acc1.w);

  const size_t o = (size_t)node * V4N + lane;
  float4 po = ((const float4*)outAcc)[o];
  po.x += wl * t.x; po.y += wl * t.y; po.z += wl * t.z; po.w += wl * t.w;
  ((float4*)outAcc)[o] = po;

  if (write_next) {
    float4 ns; ns.x = s * t.x; ns.y = s * t.y; ns.z = s * t.z; ns.w = s * t.w;
    ((float4*)Snext)[o] = ns;
  }
}

extern "C" void kernel_launch(void* const* d_in, const int* in_sizes, int n_in,
                              void* d_out, int out_size, void* d_ws, size_t ws_size,
                              hipStream_t stream) {
  (void)n_in; (void)out_size; (void)ws_size;

  const float* usr_x = (const float*)d_in[0];
  const float* itm_x = (const float*)d_in[1];
  const int*   eu    = (const int*)d_in[2];        // usr_edge_index: row0 = user ids
  const int n_usr = in_sizes[0] / D_FEAT;
  const int n_itm = in_sizes[1] / D_FEAT;
  const int E     = in_sizes[2] / 2;
  const int* ei   = eu + E;                        // row1 = item ids (itm_edge_index is the reverse; unused)

  const size_t nU = (size_t)n_usr * D_FEAT;
  const size_t nI = (size_t)n_itm * D_FEAT;

  float* Su  = (float*)d_ws;        // user-side scaled features (in-place ping)
  float* Si0 = Su  + nU;            // item-side scaled features, buffer A
  float* Si1 = Si0 + nI;            // item-side scaled features, buffer B
  u32* degu = (u32*)(Si1 + nI);     // degree counts -> inv (float) in place
  u32* degi = degu + n_usr;
  u32* rowu = degi + n_itm;         // CSR row ptr, users as dst (neighbors = items)
  u32* rowi = rowu + (n_usr + 1);   // CSR row ptr, items as dst (neighbors = users)
  u32* curu = rowi + (n_itm + 1);   // placement cursors
  u32* curi = curu + n_usr;
  u32* colu = curi + n_itm;         // item neighbors of each user, E (+4 pad)
  u32* coli = colu + (E + 4);       // user neighbors of each item, E (+4 pad)
  float* invu = (float*)degu;
  float* invi = (float*)degi;

  float* out_u = (float*)d_out;
  float* out_i = out_u + nU;

  const int nd = n_usr + n_itm;
  const int zg = (nd + 255) / 256;
  const int eg = (E + 255) / 256;

  k_zero_u32<<<zg, 256, 0, stream>>>(degu, nd);
  k_zero_u32<<<zg, 256, 0, stream>>>(curu, nd);
  k_count_deg<<<eg, 256, 0, stream>>>(eu, ei, E, degu, degi);
  k_scan_excl<<<1, 1024, 0, stream>>>(degu, rowu, n_usr);
  k_scan_excl<<<1, 1024, 0, stream>>>(degi, rowi, n_itm);
  k_deg_to_inv<<<zg, 256, 0, stream>>>(degu, nd);
  k_build_csr<<<eg, 256, 0, stream>>>(eu, ei, E, rowu, rowi, curu, curi, colu, coli);

  const int vgU = (n_usr * V4N + 255) / 256;
  const int vgI = (n_itm * V4N + 255) / 256;
  k_init<<<vgU, 256, 0, stream>>>(usr_x, invu, Su,  out_u, n_usr * V4N);
  k_init<<<vgI, 256, 0, stream>>>(itm_x, invi, Si0, out_i, n_itm * V4N);

  const float w = 0.25f;
  const int gU = (n_usr + WPB - 1) / WPB;
  const int gI = (n_itm + WPB - 1) / WPB;

  // Each layer: items-gather first (reads Su from previous layer), then
  // users-gather (reads item S from previous layer, overwrites Su in place).
  // Layer 1
  k_gather<<<gI, 256, 0, stream>>>(rowi, coli, Su,  invi, out_i, Si1, n_itm, w, 1);
  k_gather<<<gU, 256, 0, stream>>>(rowu, colu, Si0, invu, out_u, Su,  n_usr, w, 1);
  // Layer 2
  k_gather<<<gI, 256, 0, stream>>>(rowi, coli, Su,  invi, out_i, Si0, n_itm, w, 1);
  k_gather<<<gU, 256, 0, stream>>>(rowu, colu, Si1, invu, out_u, Su,  n_usr, w, 1);
  // Layer 3 (no next-layer sources needed)
  k_gather<<<gI, 256, 0, stream>>>(rowi, coli, Su,  invi, out_i, Si1, n_itm, w, 0);
  k_gather<<<gU, 256, 0, stream>>>(rowu, colu, Si0, invu, out_u, Su,  n_usr, w, 0);
}